// MyBuddyModel_64398739636359
// MI455X (gfx1250) — compile-verified
//
#include <hip/hip_runtime.h>

typedef float v2f __attribute__((ext_vector_type(2)));
typedef float v8f __attribute__((ext_vector_type(8)));

namespace {
constexpr int B = 32, T = 512, DB = 256, DP = 20, H = 192, NH = 4, HD = 48;
constexpr int F = 4096, DC = 256, MEL = 80;
constexpr int QKVW = 3 * H; // 576
constexpr float EPS = 1e-5f;
constexpr float LN1E4 = 9.2103403719761836f; // ln(10000)
}

// ---------------- WMMA helpers (CDNA5 fp32 path: V_WMMA_F32_16X16X4_F32) ----
// C/D 16x16 f32 layout: lane = {half = lane>>4, j = lane&15}; element (m,n):
//   m = r + 8*half (r = acc component 0..7), n = j.
// A 16x4 f32: lane r=lane&15 holds A[r][2*half + {0,1}].
// B 4x16 f32 (as B^T rows): lane r holds B[2*half + {0,1}][r]; with Bt stored
// K-contiguous this is a float2 at Bt[r*ldb + 2*half].

__device__ __forceinline__ v8f wmma4(v2f a, v2f b, v8f c) {
  return __builtin_amdgcn_wmma_f32_16x16x4_f32(
      false, a, false, b, (short)0, c, false, false);
}

// C(16x16) += A(16xK, row-major, lda) * Bt(16xK, row-major, ldb)^T
__device__ __forceinline__ v8f gemm_kk(const float* __restrict__ A, long lda,
                                       const float* __restrict__ Bt, long ldb,
                                       int K, int lane) {
  const int half = lane >> 4, r = lane & 15;
  const float* ap = A + (long)r * lda + 2 * half;
  const float* bp = Bt + (long)r * ldb + 2 * half;
  v8f acc = {0.f, 0.f, 0.f, 0.f, 0.f, 0.f, 0.f, 0.f};
  for (int k = 0; k < K; k += 4) {
    v2f a = *(const v2f*)(ap + k);
    v2f b = *(const v2f*)(bp + k);
    acc = wmma4(a, b, acc);
  }
  return acc;
}

// ---------------- Kernels ---------------------------------------------------

// x = bert @ in_w^T + in_b + pros @ pros_w^T + pros_b + emo[ids] + sinusoid
__global__ void k_embed(const float* __restrict__ bert,
                        const float* __restrict__ pros,
                        const float* __restrict__ in_w, const float* __restrict__ in_b,
                        const float* __restrict__ pros_w, const float* __restrict__ pros_b,
                        const float* __restrict__ emo, const int* __restrict__ emo_ids,
                        float* __restrict__ x) {
  const int lane = threadIdx.x & 31;
  const int wave = blockIdx.x * (blockDim.x >> 5) + (threadIdx.x >> 5);
  const int ntn = H / 16; // 12
  const int tn = wave % ntn, tm = wave / ntn;
  if (tm >= (B * T) / 16) return;
  const int m0 = tm * 16, n0 = tn * 16;
  v8f acc = gemm_kk(bert + (long)m0 * DB, DB, in_w + (long)n0 * DB, DB, DB, lane);
  const int j = lane & 15, half = lane >> 4;
  const int n = n0 + j;
  const float bias = in_b[n] + pros_b[n];
  const float* pw = pros_w + (long)n * DP;
  const int i2 = n >> 1;
  const float div = __expf((float)(2 * i2) * (-LN1E4 / (float)H));
  for (int r = 0; r < 8; ++r) {
    const int m = m0 + r + 8 * half;
    const int b = m / T, t = m % T;
    float s = acc[r] + bias;
    const float* pr = pros + (long)m * DP;
    for (int k = 0; k < DP; ++k) s += pr[k] * pw[k];
    s += emo[emo_ids[b] * H + n];
    const float ang = (float)t * div;
    s += (n & 1) ? __cosf(ang) : __sinf(ang);
    x[(long)m * H + n] = s;
  }
}

// wave-per-row LayerNorm over H=192
__global__ void k_layernorm(const float* __restrict__ x, const float* __restrict__ g,
                            const float* __restrict__ bt, float* __restrict__ y, int rows) {
  const int lane = threadIdx.x & 31;
  const int wave = blockIdx.x * (blockDim.x >> 5) + (threadIdx.x >> 5);
  if (wave >= rows) return;
  const float* xr = x + (long)wave * H;
  float v[6], s = 0.f, s2 = 0.f;
  for (int i = 0; i < 6; ++i) {
    float t = xr[lane + 32 * i];
    v[i] = t; s += t; s2 += t * t;
  }
  for (int off = 16; off; off >>= 1) {
    s += __shfl_xor(s, off, 32);
    s2 += __shfl_xor(s2, off, 32);
  }
  const float mean = s * (1.f / H);
  const float rstd = rsqrtf(s2 * (1.f / H) - mean * mean + EPS);
  float* yr = y + (long)wave * H;
  for (int i = 0; i < 6; ++i) {
    const int c = lane + 32 * i;
    yr[c] = (v[i] - mean) * rstd * g[c] + bt[c];
  }
}

// Generic C[MxN] = A[MxK] @ W[NxK]^T + bias, optional accumulate into C
__global__ void k_gemm(const float* __restrict__ A, const float* __restrict__ W,
                       const float* __restrict__ bias, float* __restrict__ C,
                       int M, int N, int K, int acc_add) {
  const int lane = threadIdx.x & 31;
  const int wave = blockIdx.x * (blockDim.x >> 5) + (threadIdx.x >> 5);
  const int ntn = N >> 4;
  const int tn = wave % ntn, tm = wave / ntn;
  if (tm >= (M >> 4)) return;
  v8f acc = gemm_kk(A + (long)tm * 16 * K, K, W + (long)tn * 16 * K, K, K, lane);
  const int j = lane & 15, half = lane >> 4;
  const int n = tn * 16 + j;
  const float bv = bias ? bias[n] : 0.f;
  float* Cp = C + (long)tm * 16 * N + n;
  for (int r = 0; r < 8; ++r) {
    const long idx = (long)(r + 8 * half) * N;
    float s = acc[r] + bv;
    if (acc_add) s += Cp[idx];
    Cp[idx] = s;
  }
}

// attention scores: att[b,h] = (Q @ K^T) / sqrt(48)
__global__ void k_scores(const float* __restrict__ qkv, float* __restrict__ att) {
  const int lane = threadIdx.x & 31;
  int wave = blockIdx.x * (blockDim.x >> 5) + (threadIdx.x >> 5);
  const int tn = wave & 31; wave >>= 5;
  const int tm = wave & 31; wave >>= 5;
  const int h = wave & 3;  const int b = wave >> 2;
  if (b >= B) return;
  const float* qb = qkv + (long)b * T * QKVW + h * HD;
  const float* kb = qb + H;
  v8f acc = gemm_kk(qb + (long)tm * 16 * QKVW, QKVW,
                    kb + (long)tn * 16 * QKVW, QKVW, HD, lane);
  const int j = lane & 15, half = lane >> 4;
  float* Cp = att + ((long)(b * NH + h) * T + tm * 16) * T + (long)tn * 16 + j;
  const float scale = 0.14433756729740643f; // 1/sqrt(48)
  for (int r = 0; r < 8; ++r) Cp[(long)(r + 8 * half) * T] = acc[r] * scale;
}

// wave-per-row softmax over 512 columns
__global__ void k_softmax(float* __restrict__ att, int rows) {
  const int lane = threadIdx.x & 31;
  const int wave = blockIdx.x * (blockDim.x >> 5) + (threadIdx.x >> 5);
  if (wave >= rows) return;
  float* rp = att + (long)wave * T;
  float v[16], mx = -3.4e38f;
  for (int i = 0; i < 16; ++i) { v[i] = rp[lane + 32 * i]; mx = fmaxf(mx, v[i]); }
  for (int off = 16; off; off >>= 1) mx = fmaxf(mx, __shfl_xor(mx, off, 32));
  float s = 0.f;
  for (int i = 0; i < 16; ++i) { v[i] = __expf(v[i] - mx); s += v[i]; }
  for (int off = 16; off; off >>= 1) s += __shfl_xor(s, off, 32);
  const float inv = 1.f / s;
  for (int i = 0; i < 16; ++i) rp[lane + 32 * i] = v[i] * inv;
}

// o[b,t,h*48+d] = att[b,h] @ V[b,h]   (K = T = 512; V accessed k-strided)
__global__ void k_att_o(const float* __restrict__ qkv, const float* __restrict__ att,
                        float* __restrict__ obuf) {
  const int lane = threadIdx.x & 31;
  int wave = blockIdx.x * (blockDim.x >> 5) + (threadIdx.x >> 5);
  const int tn = wave % 3; wave /= 3;
  const int tm = wave & 31; wave >>= 5;
  const int h = wave & 3;  const int b = wave >> 2;
  if (b >= B) return;
  const float* A = att + (long)(b * NH + h) * T * T + (long)tm * 16 * T;
  const float* Vp = qkv + (long)b * T * QKVW + 2 * H + h * HD + tn * 16;
  const int half = lane >> 4, r = lane & 15;
  const float* ap = A + (long)r * T + 2 * half;
  const float* bp = Vp + r + (long)(2 * half) * QKVW;
  v8f acc = {0.f, 0.f, 0.f, 0.f, 0.f, 0.f, 0.f, 0.f};
  for (int k = 0; k < T; k += 4) {
    v2f a = *(const v2f*)(ap + k);
    v2f bb;
    bb.x = bp[(long)k * QKVW];
    bb.y = bp[(long)k * QKVW + QKVW];
    acc = wmma4(a, bb, acc);
  }
  const int j = lane & 15;
  float* Cp = obuf + (long)(b * T + tm * 16) * H + h * HD + tn * 16 + j;
  for (int rr = 0; rr < 8; ++rr) Cp[(long)(rr + 8 * half) * H] = acc[rr];
}

// depthwise conv k=5 over T + SiLU
__global__ void k_dwconv(const float* __restrict__ x, const float* __restrict__ w,
                         const float* __restrict__ bias, float* __restrict__ y) {
  const long idx = (long)blockIdx.x * blockDim.x + threadIdx.x;
  if (idx >= (long)B * T * H) return;
  const int c = (int)(idx % H);
  const long bt = idx / H;
  const int t = (int)(bt % T);
  const long b = bt / T;
  float s = bias[c];
  for (int j = 0; j < 5; ++j) {
    const int tt = t + j - 2;
    if (tt >= 0 && tt < T) s += x[(b * T + tt) * H + c] * w[c * 5 + j];
  }
  y[idx] = s / (1.f + __expf(-s)); // SiLU
}

// per-batch inclusive cumsum of durations
__global__ void k_cumsum(const int* __restrict__ dur, int* __restrict__ cum) {
  const int b = threadIdx.x;
  if (b >= B) return;
  int s = 0;
  for (int t = 0; t < T; ++t) { s += dur[b * T + t]; cum[b * T + t] = s; }
}

// length regulator: xe[b,f,:] = x[b, searchsorted(cum,f,'right'), :] * valid
__global__ void k_lenreg(const float* __restrict__ x, const int* __restrict__ cum,
                         float* __restrict__ xe) {
  const int lane = threadIdx.x & 31;
  const int wave = blockIdx.x * (blockDim.x >> 5) + (threadIdx.x >> 5);
  if (wave >= B * F) return;
  const int b = wave / F, f = wave % F;
  const int* c = cum + b * T;
  int lo = 0, hi = T;
  while (lo < hi) { const int mid = (lo + hi) >> 1; if (c[mid] <= f) lo = mid + 1; else hi = mid; }
  const int tok = lo < (T - 1) ? lo : (T - 1);
  const float scale = (f < c[T - 1]) ? 1.f : 0.f;
  const float* src = x + ((long)b * T + tok) * H;
  float* dst = xe + (long)wave * H;
  for (int i = 0; i < 6; ++i) dst[lane + 32 * i] = src[lane + 32 * i] * scale;
}

// repack conv weights (O,I,3) -> (3,O,I) so B-fragments are K-contiguous
__global__ void k_repack3(const float* __restrict__ w, float* __restrict__ wt) {
  const long i = (long)blockIdx.x * blockDim.x + threadIdx.x;
  if (i >= 3L * DC * DC) return;
  const int j = (int)(i / (DC * DC));
  const int rem = (int)(i % (DC * DC));
  const int co = rem / DC, ci = rem % DC;
  wt[i] = w[(long)(co * DC + ci) * 3 + j];
}

// dilated k=3 conv as implicit WMMA GEMM (K = 3*256), zero-pad at frame edges
__global__ void k_conv3(const float* __restrict__ in, const float* __restrict__ wt,
                        const float* __restrict__ bias, float* __restrict__ out, int dil) {
  const int lane = threadIdx.x & 31;
  const int wave = blockIdx.x * (blockDim.x >> 5) + (threadIdx.x >> 5);
  const int tn = wave & 15, tm = wave >> 4;
  if (tm >= (B * F) / 16) return;
  const int half = lane >> 4, r = lane & 15;
  const int m = tm * 16 + r;
  const int b = m >> 12;      // F = 4096
  const int f = m & (F - 1);
  v8f acc = {0.f, 0.f, 0.f, 0.f, 0.f, 0.f, 0.f, 0.f};
  const float* wrow = wt + (long)(tn * 16 + r) * DC + 2 * half;
  for (int j = 0; j < 3; ++j) {
    const int ff = f + (j - 1) * dil;
    const bool ok = ((unsigned)ff < (unsigned)F);
    const float* ap = in + ((long)(b << 12) + (ok ? ff : 0)) * DC + 2 * half;
    const float* bp = wrow + (long)j * DC * DC;
    for (int k = 0; k < DC; k += 4) {
      v2f a;
      if (ok) a = *(const v2f*)(ap + k); else { a.x = 0.f; a.y = 0.f; }
      v2f bb = *(const v2f*)(bp + k);
      acc = wmma4(a, bb, acc);
    }
  }
  const int j2 = lane & 15, n = tn * 16 + j2;
  const float bv = bias[n];
  float* Cp = out + (long)tm * 16 * DC + n;
  for (int rr = 0; rr < 8; ++rr) Cp[(long)(rr + 8 * half) * DC] = acc[rr] + bv;
}

// group-norm statistics: one block per (b, g); 32 channels x 4096 frames
__global__ void k_gnstats(const float* __restrict__ x, float* __restrict__ stats) {
  const int bg = blockIdx.x;
  const int b = bg >> 3, g = bg & 7;
  __shared__ float sh[512];
  float s = 0.f, s2 = 0.f;
  const float* base = x + (long)b * F * DC + g * 32;
  for (int i = threadIdx.x; i < F * 32; i += blockDim.x) {
    const int f = i >> 5, c = i & 31;
    const float t = base[(long)f * DC + c];
    s += t; s2 += t * t;
  }
  sh[threadIdx.x] = s; sh[256 + threadIdx.x] = s2;
  __syncthreads();
  for (int off = 128; off; off >>= 1) {
    if ((int)threadIdx.x < off) {
      sh[threadIdx.x] += sh[threadIdx.x + off];
      sh[256 + threadIdx.x] += sh[256 + threadIdx.x + off];
    }
    __syncthreads();
  }
  if (threadIdx.x == 0) {
    const float inv = 1.f / (float)(F * 32);
    const float mean = sh[0] * inv;
    stats[bg * 2 + 0] = mean;
    stats[bg * 2 + 1] = rsqrtf(sh[256] * inv - mean * mean + EPS);
  }
}

// apply group norm (+ optional SiLU, + optional residual)
__global__ void k_gnapply(const float* __restrict__ x, const float* __restrict__ stats,
                          const float* __restrict__ gamma, const float* __restrict__ beta,
                          const float* __restrict__ res, float* __restrict__ y, int do_silu) {
  const long i = (long)blockIdx.x * blockDim.x + threadIdx.x;
  if (i >= (long)B * F * DC) return;
  const int c = (int)(i % DC);
  const int b = (int)((i / DC) / F);
  const int g = c >> 5;
  const float mean = stats[(b * 8 + g) * 2];
  const float rstd = stats[(b * 8 + g) * 2 + 1];
  float t = (x[i] - mean) * rstd * gamma[c] + beta[c];
  if (do_silu) t = t / (1.f + __expf(-t));
  if (res) t += res[i];
  y[i] = t;
}

// ---------------- Host launcher --------------------------------------------

extern "C" void kernel_launch(void* const* d_in, const int* in_sizes, int n_in,
                              void* d_out, int out_size, void* d_ws, size_t ws_size,
                              hipStream_t stream) {
  (void)in_sizes; (void)n_in; (void)out_size; (void)ws_size;
  const float* bert = (const float*)d_in[0];
  const float* pros = (const float*)d_in[1];
  const float* P    = (const float*)d_in[2];
  const int* emo_ids = (const int*)d_in[3];
  const int* dur     = (const int*)d_in[4];

  // params flattened as a JAX pytree (dict keys alphabetical, lists in order)
  size_t off = 0;
  auto take = [&](size_t n) { const float* p = P + off; off += n; return p; };
  struct DecBlk { const float *c1_b, *c1_w, *c2_b, *c2_w, *gn1_b, *gn1_g, *gn2_b, *gn2_g; } blk[4];
  for (int i = 0; i < 4; ++i) {
    blk[i].c1_b = take(DC); blk[i].c1_w = take((size_t)DC * DC * 3);
    blk[i].c2_b = take(DC); blk[i].c2_w = take((size_t)DC * DC * 3);
    blk[i].gn1_b = take(DC); blk[i].gn1_g = take(DC);
    blk[i].gn2_b = take(DC); blk[i].gn2_g = take(DC);
  }
  const float* dec_in_b  = take(DC);
  const float* dec_in_w  = take((size_t)DC * H);
  const float* dec_out_b = take(MEL);
  const float* dec_out_w = take((size_t)MEL * DC);
  const float* emo       = take(8 * H);
  struct EncL { const float *dw_b, *dw_w, *ln_b, *ln_g, *out_b, *out_w, *pw_b, *pw_w, *qkv_b, *qkv_w; } enc[2];
  for (int i = 0; i < 2; ++i) {
    enc[i].dw_b = take(H);  enc[i].dw_w = take((size_t)H * 5);
    enc[i].ln_b = take(H);  enc[i].ln_g = take(H);
    enc[i].out_b = take(H); enc[i].out_w = take((size_t)H * H);
    enc[i].pw_b = take(H);  enc[i].pw_w = take((size_t)H * H);
    enc[i].qkv_b = take(3 * H); enc[i].qkv_w = take((size_t)3 * H * H);
  }
  const float* in_b   = take(H);
  const float* in_w   = take((size_t)H * DB);
  const float* pros_b = take(H);
  const float* pros_w = take((size_t)H * DP);

  // workspace layout (floats), regions reused phase-to-phase
  constexpr long SZ_BIG = 33554432L; // 32*4096*256 == 32*4*512*512
  float* att = (float*)d_ws;        // encoder att  -> decoder t1
  float* xe  = att + SZ_BIG;        // bridge xe    -> decoder t2
  float* R3  = xe + SZ_BIG;         // encoder scratch -> decoder xc
  float* x    = R3;
  float* nx   = R3 + 3145728L;
  float* qkvb = R3 + 2 * 3145728L;
  float* obuf = qkvb + 9437184L;
  float* xdw  = obuf + 3145728L;
  float* xc   = R3;                 // reused after length regulator
  float* small = R3 + SZ_BIG;
  int*   cum   = (int*)small;
  float* stats = small + 16384;
  float* wt    = stats + 512;
  float* t1 = att;
  float* t2 = xe;

  const dim3 blk256(256);

  // ---- encoder front-end ----
  k_embed<<<1536, blk256, 0, stream>>>(bert, pros, in_w, in_b, pros_w, pros_b,
                                       emo, emo_ids, x);

  for (int L = 0; L < 2; ++L) {
    k_layernorm<<<2048, blk256, 0, stream>>>(x, enc[L].ln_g, enc[L].ln_b, nx, B * T);
    k_gemm<<<4608, blk256, 0, stream>>>(nx, enc[L].qkv_w, enc[L].qkv_b, qkvb,
                                        B * T, QKVW, H, 0);
    k_scores<<<16384, blk256, 0, stream>>>(qkvb, att);
    k_softmax<<<8192, blk256, 0, stream>>>(att, B * NH * T);
    k_att_o<<<1536, blk256, 0, stream>>>(qkvb, att, obuf);
    k_gemm<<<1536, blk256, 0, stream>>>(obuf, enc[L].out_w, enc[L].out_b, x,
                                        B * T, H, H, 1);
    k_dwconv<<<12288, blk256, 0, stream>>>(x, enc[L].dw_w, enc[L].dw_b, xdw);
    k_gemm<<<1536, blk256, 0, stream>>>(xdw, enc[L].pw_w, enc[L].pw_b, x,
                                        B * T, H, H, 1);
  }

  // ---- length regulator ----
  k_cumsum<<<1, 32, 0, stream>>>(dur, cum);
  k_lenreg<<<16384, blk256, 0, stream>>>(x, cum, xe);

  // ---- decoder ----
  k_gemm<<<16384, blk256, 0, stream>>>(xe, dec_in_w, dec_in_b, xc,
                                       B * F, DC, H, 0);
  for (int i = 0; i < 4; ++i) {
    const int d = 1 << i;
    k_repack3<<<768, blk256, 0, stream>>>(blk[i].c1_w, wt);
    k_conv3<<<16384, blk256, 0, stream>>>(xc, wt, blk[i].c1_b, t1, d);
    k_gnstats<<<256, blk256, 0, stream>>>(t1, stats);
    k_gnapply<<<131072, blk256, 0, stream>>>(t1, stats, blk[i].gn1_g, blk[i].gn1_b,
                                             nullptr, t1, 1);
    k_repack3<<<768, blk256, 0, stream>>>(blk[i].c2_w, wt);
    k_conv3<<<16384, blk256, 0, stream>>>(t1, wt, blk[i].c2_b, t2, d);
    k_gnstats<<<256, blk256, 0, stream>>>(t2, stats);
    k_gnapply<<<131072, blk256, 0, stream>>>(t2, stats, blk[i].gn2_g, blk[i].gn2_b,
                                             xc, xc, 0);
  }
  k_gemm<<<5120, blk256, 0, stream>>>(xc, dec_out_w, dec_out_b, (float*)d_out,
                                      B * F, MEL, DC, 0);
}